// GATModule_88364657148499
// MI455X (gfx1250) — compile-verified
//
#include <hip/hip_runtime.h>
#include <hip/hip_bf16.h>

typedef __attribute__((ext_vector_type(2))) float v2f;
typedef __attribute__((ext_vector_type(8))) float v8f;

#define HEADS 4
#define CPH 32           // channels per head
#define DIMF 128         // total feature dim
#define NEG_SLOPE 0.2f

// ---------------------------------------------------------------------------
// Kernel 1: x_proj = x @ W  via V_WMMA_F32_16X16X4_F32 (full fp32 precision)
// One wave (32 lanes) computes a 16-row x 128-col strip.
// A (16x4 f32): lane L(0..15) holds A[M=L][k0], A[M=L][k0+1] in v0,v1 with
//               k0 = 0 for lanes 0-15, k0 = 2 for lanes 16-31.
// B (4x16 f32): lane L holds B[k0][N=L&15], B[k0+1][N=L&15].
// D (16x16 f32): vgpr v holds D[M = v + 8*(L>>4)][N = L&15].
// ---------------------------------------------------------------------------
__global__ void __launch_bounds__(256) gat_gemm_wmma(const float* __restrict__ x,
                                                     const float* __restrict__ W,
                                                     float* __restrict__ xproj,
                                                     int N) {
  const int lane  = threadIdx.x & 31;
  const int wave  = threadIdx.x >> 5;
  const int strip = blockIdx.x * 8 + wave;
  const int nStrips = (N + 15) >> 4;
  if (strip >= nStrips) return;

  const int m0    = strip << 4;
  const int mrow  = m0 + (lane & 15);
  const int mrd   = (mrow < N) ? mrow : (N - 1);   // clamp reads on tail strip
  const int khalf = (lane >> 4) << 1;              // 0 or 2
  const int ncl   = lane & 15;

  // Preload all A fragments for this strip (K = 128 -> 32 steps of 4)
  v2f afrag[32];
  const float* xrow = x + (size_t)mrd * DIMF + khalf;
#pragma unroll
  for (int kk = 0; kk < 32; ++kk) {
    afrag[kk].x = xrow[kk * 4 + 0];
    afrag[kk].y = xrow[kk * 4 + 1];
  }

#pragma unroll
  for (int nt = 0; nt < 8; ++nt) {
    const int n0 = nt << 4;
    v8f acc = {};
    const float* wbase = W + (size_t)khalf * DIMF + n0 + ncl;
#pragma unroll
    for (int kk = 0; kk < 32; ++kk) {
      v2f b;
      b.x = wbase[(size_t)(kk * 4 + 0) * DIMF];
      b.y = wbase[(size_t)(kk * 4 + 1) * DIMF];
      acc = __builtin_amdgcn_wmma_f32_16x16x4_f32(
          /*neg_a=*/false, afrag[kk], /*neg_b=*/false, b,
          /*c_mod=*/(short)0, acc, /*reuse_a=*/false, /*reuse_b=*/false);
    }
#pragma unroll
    for (int v = 0; v < 8; ++v) {
      const int row = m0 + v + ((lane >> 4) << 3);
      if (row < N) xproj[(size_t)row * DIMF + n0 + ncl] = acc[v];
    }
  }
}

// ---------------------------------------------------------------------------
// Kernel 2: per-(node,head) attention coefficients
// ---------------------------------------------------------------------------
__global__ void __launch_bounds__(256) attn_coef(const float* __restrict__ xproj,
                                                 const float* __restrict__ att_src,
                                                 const float* __restrict__ att_dst,
                                                 float* __restrict__ a_src,
                                                 float* __restrict__ a_dst,
                                                 int N) {
  const int t = blockIdx.x * blockDim.x + threadIdx.x;
  if (t >= N * HEADS) return;
  const int h = t & (HEADS - 1);
  const int n = t >> 2;
  const float* xp = xproj + (size_t)n * DIMF + h * CPH;
  const float* as = att_src + h * CPH;
  const float* ad = att_dst + h * CPH;
  float ss = 0.f, dd = 0.f;
#pragma unroll
  for (int c = 0; c < CPH; ++c) {
    const float v = xp[c];
    ss += v * as[c];
    dd += v * ad[c];
  }
  a_src[t] = ss;
  a_dst[t] = dd;
}

// order-preserving float -> uint encoding (for atomic max over floats)
__device__ __forceinline__ unsigned enc_f(float f) {
  unsigned u = __float_as_uint(f);
  return (u & 0x80000000u) ? ~u : (u | 0x80000000u);
}
__device__ __forceinline__ float dec_f(unsigned e) {
  unsigned u = (e & 0x80000000u) ? (e & 0x7FFFFFFFu) : ~e;
  return __uint_as_float(u);
}
__device__ __forceinline__ float lrelu(float v) {
  return v >= 0.f ? v : NEG_SLOPE * v;
}

// ---------------------------------------------------------------------------
// Kernel 3: per-edge logits + segment max (encoded atomicMax)
// ---------------------------------------------------------------------------
__global__ void __launch_bounds__(256) edge_alpha(const int* __restrict__ ei,
                                                  int E, int E2,
                                                  const float* __restrict__ a_src,
                                                  const float* __restrict__ a_dst,
                                                  float* __restrict__ alpha,
                                                  unsigned* __restrict__ m) {
  const int e = blockIdx.x * blockDim.x + threadIdx.x;
  if (e >= E2) return;
  int s, d;
  if (e < E) { s = ei[e]; d = ei[E + e]; } else { s = d = e - E; }
  const float4 as = *(const float4*)(a_src + (size_t)s * HEADS);
  const float4 ad = *(const float4*)(a_dst + (size_t)d * HEADS);
  float4 al;
  al.x = lrelu(as.x + ad.x);
  al.y = lrelu(as.y + ad.y);
  al.z = lrelu(as.z + ad.z);
  al.w = lrelu(as.w + ad.w);
  *(float4*)(alpha + (size_t)e * HEADS) = al;
  unsigned* mb = m + (size_t)d * HEADS;
  atomicMax(mb + 0, enc_f(al.x));
  atomicMax(mb + 1, enc_f(al.y));
  atomicMax(mb + 2, enc_f(al.z));
  atomicMax(mb + 3, enc_f(al.w));
}

// ---------------------------------------------------------------------------
// Kernel 4: ex = exp(alpha - m[dst]); segment sum (atomicAdd)
// ---------------------------------------------------------------------------
__global__ void __launch_bounds__(256) edge_exp(const int* __restrict__ ei,
                                                int E, int E2,
                                                const unsigned* __restrict__ m,
                                                float* __restrict__ alpha,
                                                float* __restrict__ ssum) {
  const int e = blockIdx.x * blockDim.x + threadIdx.x;
  if (e >= E2) return;
  const int d = (e < E) ? ei[E + e] : (e - E);
  float4 al = *(const float4*)(alpha + (size_t)e * HEADS);
  const unsigned* mb = m + (size_t)d * HEADS;
  al.x = __expf(al.x - dec_f(mb[0]));
  al.y = __expf(al.y - dec_f(mb[1]));
  al.z = __expf(al.z - dec_f(mb[2]));
  al.w = __expf(al.w - dec_f(mb[3]));
  *(float4*)(alpha + (size_t)e * HEADS) = al;
  float* sb = ssum + (size_t)d * HEADS;
  unsafeAtomicAdd(sb + 0, al.x);
  unsafeAtomicAdd(sb + 1, al.y);
  unsafeAtomicAdd(sb + 2, al.z);
  unsafeAtomicAdd(sb + 3, al.w);
}

// ---------------------------------------------------------------------------
// Kernel 5: wave-per-edge weighted scatter-aggregate (fp32 hw atomics)
// lane L handles channels [4L, 4L+4); head h = L >> 3
// ---------------------------------------------------------------------------
__global__ void __launch_bounds__(256) edge_aggr(const int* __restrict__ ei,
                                                 int E, int E2,
                                                 const float* __restrict__ ex,
                                                 const float* __restrict__ ssum,
                                                 const float* __restrict__ xproj,
                                                 float* __restrict__ out) {
  const int gw   = (int)(((size_t)blockIdx.x * blockDim.x + threadIdx.x) >> 5);
  const int lane = threadIdx.x & 31;
  if (gw >= E2) return;
  int s, d;
  if (gw < E) { s = ei[gw]; d = ei[E + gw]; } else { s = d = gw - E; }
  const int c = lane << 2;
  const int h = lane >> 3;
  const float a = ex[(size_t)gw * HEADS + h] /
                  (ssum[(size_t)d * HEADS + h] + 1e-16f);
  const float4 v = *(const float4*)(xproj + (size_t)s * DIMF + c);
  float* o = out + (size_t)d * DIMF + c;
  unsafeAtomicAdd(o + 0, a * v.x);
  unsafeAtomicAdd(o + 1, a * v.y);
  unsafeAtomicAdd(o + 2, a * v.z);
  unsafeAtomicAdd(o + 3, a * v.w);
}

// ---------------------------------------------------------------------------
// Kernel 6: out = relu(out + bias)
// ---------------------------------------------------------------------------
__global__ void __launch_bounds__(256) bias_relu(float* __restrict__ out,
                                                 const float* __restrict__ bias,
                                                 int total) {
  const int i = blockIdx.x * blockDim.x + threadIdx.x;
  if (i >= total) return;
  const float v = out[i] + bias[i & (DIMF - 1)];
  out[i] = v > 0.f ? v : 0.f;
}

extern "C" void kernel_launch(void* const* d_in, const int* in_sizes, int n_in,
                              void* d_out, int out_size, void* d_ws, size_t ws_size,
                              hipStream_t stream) {
  const int N  = in_sizes[0] / DIMF;
  const int E  = in_sizes[1] / 2;
  const int E2 = E + N;

  const float* x      = (const float*)d_in[0];
  const int*   ei     = (const int*)d_in[1];
  const float* W      = (const float*)d_in[2];
  const float* att_s  = (const float*)d_in[3];
  const float* att_d  = (const float*)d_in[4];
  const float* bias   = (const float*)d_in[5];
  float* out = (float*)d_out;

  // carve workspace
  char* ws = (char*)d_ws;
  float*    xproj = (float*)ws;    ws += (size_t)N * DIMF * sizeof(float);
  float*    a_src = (float*)ws;    ws += (size_t)N * HEADS * sizeof(float);
  float*    a_dst = (float*)ws;    ws += (size_t)N * HEADS * sizeof(float);
  unsigned* m     = (unsigned*)ws; ws += (size_t)N * HEADS * sizeof(unsigned);
  float*    ssum  = (float*)ws;    ws += (size_t)N * HEADS * sizeof(float);
  float*    alpha = (float*)ws;    ws += (size_t)E2 * HEADS * sizeof(float);

  hipMemsetAsync(out,  0, (size_t)N * DIMF * sizeof(float), stream);
  hipMemsetAsync(m,    0, (size_t)N * HEADS * sizeof(unsigned), stream);  // enc(-inf) never wins
  hipMemsetAsync(ssum, 0, (size_t)N * HEADS * sizeof(float), stream);

  const int nStrips = (N + 15) / 16;
  gat_gemm_wmma<<<(nStrips + 7) / 8, 256, 0, stream>>>(x, W, xproj, N);
  attn_coef<<<(N * HEADS + 255) / 256, 256, 0, stream>>>(xproj, att_s, att_d, a_src, a_dst, N);
  edge_alpha<<<(E2 + 255) / 256, 256, 0, stream>>>(ei, E, E2, a_src, a_dst, alpha, m);
  edge_exp<<<(E2 + 255) / 256, 256, 0, stream>>>(ei, E, E2, m, alpha, ssum);
  edge_aggr<<<(int)(((size_t)E2 * 32 + 255) / 256), 256, 0, stream>>>(ei, E, E2, alpha, ssum, xproj, out);
  bias_relu<<<(N * DIMF + 255) / 256, 256, 0, stream>>>(out, bias, N * DIMF);
}